// HGPSLPool_52974126629104
// MI455X (gfx1250) — compile-verified
//
#include <hip/hip_runtime.h>

// ---------------- problem constants ----------------
#define B_      32
#define N_      1024
#define D_      128
#define K_      820                     // ceil(0.8 * 1024)
#define E_      524288                  // B*N*DEG original edges
#define NNODES  32768                   // B*N
#define ETOT    (E_ + NNODES)           // edges + self loops
#define NEG_SLOPE 0.2f

// ---------------- output layout (flat floats, return order) ----------------
#define XPOOL_OFF  0LL
#define XPOOL_SZ   ((long long)B_ * K_ * D_)            // 3,358,720
#define EIDX_OFF   (XPOOL_OFF + XPOOL_SZ)
#define EIDX_HALF  ((long long)B_ * K_ * K_)            // 21,516,800
#define EIDX_SZ    (2LL * EIDX_HALF)
#define EATTR_OFF  (EIDX_OFF + EIDX_SZ)
#define BATCH_OFF  (EATTR_OFF + EIDX_HALF)

// ---------------- workspace layout (float units) ----------------
#define WS_DEG     0LL                      // 32768 floats (deg -> dinv in place)
#define WS_SCORES  32768LL                  // 32768 floats
#define WS_INFO    65536LL                  // 32768*128 floats
#define WS_S1      (WS_INFO + (long long)NNODES * D_)   // 26240 floats
#define WS_S2      (WS_S1 + (long long)B_ * K_)         // 26240 floats
#define WS_PERM    (WS_S2 + (long long)B_ * K_)         // 26240 ints
#define WS_BMAT    (WS_PERM + (long long)B_ * K_)       // 2048 floats (16 x 128, [n][k])

typedef float v2f __attribute__((ext_vector_type(2)));
typedef float v8f __attribute__((ext_vector_type(8)));

// ---------------- 1: init deg (=1 from self loop) and zero info ----------------
__global__ void k_init(float* __restrict__ deg, float* __restrict__ info) {
  long long gid = (long long)blockIdx.x * blockDim.x + threadIdx.x;
  if (gid < (long long)NNODES * D_) info[gid] = 0.0f;
  if (gid < NNODES)                 deg[gid]  = 1.0f;
}

// ---------------- 2: degree accumulation over row index ----------------
__global__ void k_degree(const int* __restrict__ row, float* __restrict__ deg) {
  int e = blockIdx.x * blockDim.x + threadIdx.x;
  if (e < E_) atomicAdd(&deg[row[e]], 1.0f);
}

// ---------------- 3: dinv = rsqrt(deg) in place ----------------
__global__ void k_dinv(float* __restrict__ deg) {
  int i = blockIdx.x * blockDim.x + threadIdx.x;
  if (i < NNODES) {
    float d = deg[i];
    deg[i] = (d > 0.0f) ? rsqrtf(d) : 0.0f;
  }
}

// ---------------- 4: info[c] += dinv[r]*dinv[c]*x[r]  (incl. self loops) -------
__global__ void k_prop(const int* __restrict__ row, const int* __restrict__ col,
                       const float* __restrict__ x, const float* __restrict__ dinv,
                       float* __restrict__ info) {
  long long gid = (long long)blockIdx.x * blockDim.x + threadIdx.x;
  long long eid = gid >> 5;            // 32 threads per edge, 4 dims each
  int chunk = (int)(gid & 31);
  if (eid >= ETOT) return;
  int r, c;
  if (eid < E_) { r = row[eid]; c = col[eid]; }
  else          { r = c = (int)(eid - E_); }
  float nrm = dinv[r] * dinv[c];
  float4 xv = *(const float4*)(x + (long long)r * D_ + chunk * 4);
  float* dst = info + (long long)c * D_ + chunk * 4;
  atomicAdd(dst + 0, nrm * xv.x);
  atomicAdd(dst + 1, nrm * xv.y);
  atomicAdd(dst + 2, nrm * xv.z);
  atomicAdd(dst + 3, nrm * xv.w);
}

// ---------------- 5: scores = sum_d |info| ----------------
__global__ void k_scores(const float* __restrict__ info, float* __restrict__ scores) {
  int i = blockIdx.x * blockDim.x + threadIdx.x;
  if (i >= NNODES) return;
  const float4* p = (const float4*)(info + (long long)i * D_);
  float s = 0.0f;
#pragma unroll 8
  for (int d = 0; d < D_ / 4; ++d) {
    float4 v = p[d];
    s += fabsf(v.x) + fabsf(v.y) + fabsf(v.z) + fabsf(v.w);
  }
  scores[i] = s;
}

// ---------------- 6: per-graph top-K via rank (stable, descending) ------------
__global__ void k_topk(const float* __restrict__ scores, int* __restrict__ perm) {
  __shared__ float s[N_];
  int b = blockIdx.x;
  int i = threadIdx.x;
  s[i] = scores[b * N_ + i];
  __syncthreads();
  float si = s[i];
  int rank = 0;
  for (int j = 0; j < N_; ++j) {
    float sj = s[j];
    rank += (sj > si) || (sj == si && j < i);
  }
  if (rank < K_) perm[b * K_ + rank] = b * N_ + i;
}

// ---------------- 7: gather x_pool + batch_pool (float4) ----------------
__global__ void k_gather(const float4* __restrict__ x4, const int* __restrict__ perm,
                         const int* __restrict__ batch, float* __restrict__ out) {
  long long gid = (long long)blockIdx.x * blockDim.x + threadIdx.x;  // float4 units
  if (gid >= XPOOL_SZ / 4) return;
  int t  = (int)(gid >> 5);          // D_/4 = 32 chunks per pooled row
  int dq = (int)(gid & 31);
  int src = perm[t];
  ((float4*)(out + XPOOL_OFF))[gid] = x4[(long long)src * 32 + dq];
  if (dq == 0) out[BATCH_OFF + t] = (float)batch[src];
}

// ---------------- 8a: build dense B matrix 16x128 ([n][k]) ----------------
// col n=0 -> a_src, n=1 -> a_dst, n>=2 -> 0
__global__ void k_bmat(const float* __restrict__ attn, float* __restrict__ bmat) {
  int idx = blockIdx.x * blockDim.x + threadIdx.x;   // 2048
  if (idx < 16 * D_) {
    int n = idx >> 7;
    int k = idx & 127;
    bmat[idx] = (n == 0) ? attn[k] : ((n == 1) ? attn[D_ + k] : 0.0f);
  }
}

// ---------------- 8b: s1/s2 = Xpool x B via f32 WMMA ----------------
// One wave per 16-row tile: D += A(16x4 f32) * B(4x16 f32), 32 steps over D=128.
// B staged in LDS with async global->LDS copies (ASYNCcnt path).
__global__ void k_attn_gemm(const float* __restrict__ xp,    // pooled x in d_out
                            const float* __restrict__ bmat,  // 16x128 [n][k]
                            float* __restrict__ s1, float* __restrict__ s2) {
  __shared__ float sb[16 * D_];                       // 8 KB
  int tid = threadIdx.x;

  // async copy bmat -> sb: 2048 floats, 256 threads x 16B x 2 issues
#pragma unroll
  for (int it = 0; it < 2; ++it) {
    int idx = (it * 256 + tid) * 4;
    unsigned lds_addr = (unsigned)(unsigned long long)(&sb[idx]);
    unsigned long long gaddr = (unsigned long long)(&bmat[idx]);
    asm volatile("global_load_async_to_lds_b128 %0, %1, off"
                 :: "v"(lds_addr), "v"(gaddr) : "memory");
  }
  asm volatile("s_wait_asynccnt 0x0" ::: "memory");
  __syncthreads();

  int wave = (blockIdx.x * blockDim.x + tid) >> 5;   // 1640 waves exactly
  int lane = tid & 31;
  int half = lane >> 4;          // K sub-pair select (A/B VGPR layout)
  int n    = lane & 15;          // A row within tile / B output column
  const float* arow = xp + (long long)(wave * 16 + n) * D_;
  const float* brow = &sb[n * D_];

  v8f c = {0.f, 0.f, 0.f, 0.f, 0.f, 0.f, 0.f, 0.f};
#pragma unroll 8
  for (int k = 0; k < D_; k += 4) {
    int ka = k + 2 * half;
    v2f a;  a.x  = arow[ka]; a.y  = arow[ka + 1];   // global_load_b64
    v2f bf; bf.x = brow[ka]; bf.y = brow[ka + 1];   // ds_load_b64
    c = __builtin_amdgcn_wmma_f32_16x16x4_f32(
        /*neg_a=*/false, a, /*neg_b=*/false, bf,
        /*c_mod=*/(short)0, c, /*reuse_a=*/false, /*reuse_b=*/false);
  }
  // D layout: VGPR v, lanes 0-15 -> row v, col=lane; lanes 16-31 -> row v+8
  int rbase = wave * 16 + half * 8;
  if (n == 0) {
#pragma unroll
    for (int v = 0; v < 8; ++v) s1[rbase + v] = c[v];
  } else if (n == 1) {
#pragma unroll
    for (int v = 0; v < 8; ++v) s2[rbase + v] = c[v];
  }
}

// ---------------- 9: row softmax + dense edge index/attr stores ---------------
__global__ void k_softmax(const float* __restrict__ s1, const float* __restrict__ s2,
                          float* __restrict__ out) {
  __shared__ float srow[K_];
  __shared__ float red[256];
  int bi = blockIdx.x;               // 0 .. B*K-1  (row = (b, i))
  int b = bi / K_;
  int i = bi - b * K_;
  int tid = threadIdx.x;

  const float* s2b = s2 + b * K_;
  for (int j = tid; j < K_; j += 256) srow[j] = s2b[j];
  __syncthreads();
  float s1i = s1[bi];

  // pass 1: row max of leaky_relu(s1i + s2[j])
  float m = -3.402823e38f;
  for (int j = tid; j < K_; j += 256) {
    float w = s1i + srow[j];
    w = (w >= 0.0f) ? w : NEG_SLOPE * w;
    m = fmaxf(m, w);
  }
  red[tid] = m; __syncthreads();
  for (int s = 128; s > 0; s >>= 1) {
    if (tid < s) red[tid] = fmaxf(red[tid], red[tid + s]);
    __syncthreads();
  }
  float rowmax = red[0]; __syncthreads();

  // pass 2: sum exp
  float acc = 0.0f;
  for (int j = tid; j < K_; j += 256) {
    float w = s1i + srow[j];
    w = (w >= 0.0f) ? w : NEG_SLOPE * w;
    acc += __expf(w - rowmax);
  }
  red[tid] = acc; __syncthreads();
  for (int s = 128; s > 0; s >>= 1) {
    if (tid < s) red[tid] += red[tid + s];
    __syncthreads();
  }
  float inv = 1.0f / red[0];

  // pass 3: float4 stores (K_ = 205 * 4, rowoff multiple of 4 -> 16B aligned)
  long long rowoff = (long long)bi * K_;
  float4* eattr = (float4*)(out + EATTR_OFF + rowoff);
  float4* erow  = (float4*)(out + EIDX_OFF + rowoff);
  float4* ecol  = (float4*)(out + EIDX_OFF + EIDX_HALF + rowoff);
  float rowid   = (float)(b * K_ + i);
  float colbase = (float)(b * K_);
  for (int q = tid; q < K_ / 4; q += 256) {
    int j = q * 4;
    float4 e, rr, cc;
    float w0 = s1i + srow[j + 0]; w0 = (w0 >= 0.f) ? w0 : NEG_SLOPE * w0;
    float w1 = s1i + srow[j + 1]; w1 = (w1 >= 0.f) ? w1 : NEG_SLOPE * w1;
    float w2 = s1i + srow[j + 2]; w2 = (w2 >= 0.f) ? w2 : NEG_SLOPE * w2;
    float w3 = s1i + srow[j + 3]; w3 = (w3 >= 0.f) ? w3 : NEG_SLOPE * w3;
    e.x = __expf(w0 - rowmax) * inv;
    e.y = __expf(w1 - rowmax) * inv;
    e.z = __expf(w2 - rowmax) * inv;
    e.w = __expf(w3 - rowmax) * inv;
    rr.x = rowid; rr.y = rowid; rr.z = rowid; rr.w = rowid;
    cc.x = colbase + (float)(j + 0);
    cc.y = colbase + (float)(j + 1);
    cc.z = colbase + (float)(j + 2);
    cc.w = colbase + (float)(j + 3);
    eattr[q] = e;
    erow[q]  = rr;
    ecol[q]  = cc;
  }
}

// ---------------- host launcher ----------------
extern "C" void kernel_launch(void* const* d_in, const int* in_sizes, int n_in,
                              void* d_out, int out_size, void* d_ws, size_t ws_size,
                              hipStream_t stream) {
  (void)in_sizes; (void)n_in; (void)out_size; (void)ws_size;
  const float* x      = (const float*)d_in[0];
  const int*   eidx   = (const int*)d_in[1];     // (2, E): row then col
  const int*   batch  = (const int*)d_in[2];
  const float* attn_w = (const float*)d_in[3];   // 256 floats
  float* out = (float*)d_out;

  float* ws   = (float*)d_ws;
  float* deg  = ws + WS_DEG;        // becomes dinv in place
  float* scr  = ws + WS_SCORES;
  float* info = ws + WS_INFO;
  float* s1   = ws + WS_S1;
  float* s2   = ws + WS_S2;
  int*   perm = (int*)(ws + WS_PERM);
  float* bmat = ws + WS_BMAT;

  const int* row = eidx;
  const int* col = eidx + E_;

  long long initN = (long long)NNODES * D_;
  k_init<<<(unsigned)((initN + 255) / 256), 256, 0, stream>>>(deg, info);
  k_degree<<<(E_ + 255) / 256, 256, 0, stream>>>(row, deg);
  k_dinv<<<(NNODES + 255) / 256, 256, 0, stream>>>(deg);
  long long propN = (long long)ETOT * 32;
  k_prop<<<(unsigned)((propN + 255) / 256), 256, 0, stream>>>(row, col, x, deg, info);
  k_scores<<<(NNODES + 255) / 256, 256, 0, stream>>>(info, scr);
  k_topk<<<B_, N_, 0, stream>>>(scr, perm);
  k_gather<<<(unsigned)((XPOOL_SZ / 4 + 255) / 256), 256, 0, stream>>>(
      (const float4*)x, perm, batch, out);
  k_bmat<<<8, 256, 0, stream>>>(attn_w, bmat);
  // 1640 waves, 8 waves (256 threads) per block -> 205 blocks exactly
  k_attn_gemm<<<205, 256, 0, stream>>>(out + XPOOL_OFF, bmat, s1, s2);
  k_softmax<<<B_ * K_, 256, 0, stream>>>(s1, s2, out);
}